// GaussianMultinomialDiffusion_5866925326636
// MI455X (gfx1250) — compile-verified
//
#include <hip/hip_runtime.h>
#include <stdint.h>

// ---------------- problem constants (from reference) ----------------
#define T_STEPS   1000
#define B_ROWS    65536
#define F_CAT     24
#define D_CAT     298
#define NUM_NUM   16
#define MODEL_W   (NUM_NUM + D_CAT)        // 314
#define LOGEPS    (-69.07755278982137f)    // logf(1e-30f)
#define TPB       128                      // rows per block == threads per block
#define PI_D      3.14159265358979323846

// 1 = single TDM tensor_load_to_lds per block (gfx1250 Tensor Data Mover)
// 0 = per-lane global_load_async_to_lds_b32 loop
#define USE_TDM 1

// LDS row stride in dwords. TDM packs rows, so stride = D_CAT = 298.
// 298 == 2 mod 4 and 149 is odd -> lane*298 mod 64 covers all 32 even banks
// exactly once across a wave32: conflict-free for the uniform-j access pattern.
#define LSTRIDE   D_CAT

__constant__ int c_K[F_CAT] = {2,2,3,4,5,6,8,10,10,12,16,16,20,24,32,40,2,3,4,5,6,8,10,50};

__device__ __forceinline__ float lae(float a, float b) {   // log(exp(a)+exp(b))
    float m = fmaxf(a, b);
    return m + __logf(__expf(a - m) + __expf(b - m));
}

// ---------------- kernel 1: diffusion tables + kl_prior into d_ws ----------------
// ws layout (floats): [0,1000)=LOG_ALPHA [1000,2000)=LOG_1M_ALPHA
//                     [2000,3000)=LOG_CA [3000,4000)=LOG_1M_CA  [4000]=kl_prior
__global__ void gmd_init_tables(float* __restrict__ ws) {
    __shared__ double la_s[T_STEPS];
    __shared__ double lca_s[T_STEPS];
    const int i = threadIdx.x;
    if (i < T_STEPS) {
        double u0 = (double)i       / (double)T_STEPS;
        double u1 = (double)(i + 1) / (double)T_STEPS;
        double cA = cos((u0 + 0.008) / 1.008 * PI_D * 0.5);
        double cB = cos((u1 + 0.008) / 1.008 * PI_D * 0.5);
        double beta = 1.0 - (cB * cB) / (cA * cA);
        if (beta > 0.2) beta = 0.2;
        double la = log(1.0 - beta);
        la_s[i] = la;
        ws[i]            = (float)la;          // LOG_ALPHA
        ws[T_STEPS + i]  = (float)log(beta);   // LOG_1M_ALPHA = log(-expm1(la))
    }
    __syncthreads();
    if (i == 0) {                               // sequential cumsum, 1000 adds
        double run = 0.0;
        for (int k = 0; k < T_STEPS; ++k) { run += la_s[k]; lca_s[k] = run; }
    }
    __syncthreads();
    if (i < T_STEPS) {
        double lca = lca_s[i];
        ws[2 * T_STEPS + i] = (float)lca;                 // LOG_CA
        ws[3 * T_STEPS + i] = (float)log(-expm1(lca));    // LOG_1M_CA
    }
    __syncthreads();
    if (i == 0) {
        // kl_prior uses t = T-1 for every row -> a single constant
        float lcaT   = (float)lca_s[T_STEPS - 1];
        float l1mcaT = (float)log(-expm1(lca_s[T_STEPS - 1]));
        float klp = 0.f;
        for (int f = 0; f < F_CAT; ++f) {
            int K = c_K[f];
            float logK = __logf((float)K);
            float a = lae(lcaT,          l1mcaT - logK);  // true column
            float b = lae(LOGEPS + lcaT, l1mcaT - logK);  // other K-1 columns
            klp += __expf(a) * (a + logK) + (float)(K - 1) * __expf(b) * (b + logK);
        }
        ws[4 * T_STEPS] = klp;
    }
}

// ---------------- kernel 2: per-row loss ----------------
__global__ void __launch_bounds__(TPB)
gmd_loss_kernel(const int*   __restrict__ x_cat_idx,
                const int*   __restrict__ t_arr,
                const float* __restrict__ noise,
                const float* __restrict__ u_cat,
                const float* __restrict__ model_out,
                const float* __restrict__ tab,
                float*       __restrict__ out)
{
    extern __shared__ float smem[];            // TPB * LSTRIDE floats
    const int tid  = threadIdx.x;
    const int row0 = blockIdx.x * TPB;

    // ---- stage model_out categorical logits for this block's TPB rows into LDS ----
#if USE_TDM
    // One Tensor-Data-Mover DMA per block: 2D tile, 128 rows x 298 dwords,
    // row-major source with stride 314 dwords. 2D tensor -> descriptor groups
    // 0 and 1 only (VADDR2/3 = NULL form). workgroup_mask=0 (not in a cluster).
    if (tid == 0) {
        typedef unsigned int v4u __attribute__((ext_vector_type(4)));
        typedef unsigned int v8u __attribute__((ext_vector_type(8)));
        const uint64_t gaddr =
            (uint64_t)(uintptr_t)(model_out + (size_t)row0 * MODEL_W + NUM_NUM);
        const uint32_t laddr = (uint32_t)(uintptr_t)smem;   // LDS byte offset

        v4u g0;
        g0[0] = 1u;                                      // count=1, user descriptor
        g0[1] = laddr;                                   // lds_addr[31:0]
        g0[2] = (uint32_t)(gaddr & 0xFFFFFFFFu);         // global_addr[31:0]
        g0[3] = (uint32_t)((gaddr >> 32) & 0x1FFFFFFu)   // global_addr[56:32]
              | (2u << 30);                              // type = 2 ("image")

        v8u g1;
        g1[0] = 2u << 16;                                // data_size=2 (4 bytes)
        g1[1] = ((uint32_t)D_CAT & 0xFFFFu) << 16;       // tensor_dim0[15:0]=298
        g1[2] = ((uint32_t)TPB   & 0xFFFFu) << 16;       // dim0[31:16]=0 | tensor_dim1=128
        g1[3] = ((uint32_t)D_CAT) << 16;                 // dim1[31:16]=0 | tile_dim0=298
        g1[4] = (uint32_t)TPB;                           // tile_dim1=128 | tile_dim2=0
        g1[5] = (uint32_t)MODEL_W;                       // tensor_dim0_stride[31:0]=314
        g1[6] = 0u;                                      // stride[47:32] | dim1_stride[15:0]
        g1[7] = 0u;                                      // tensor_dim1_stride[47:16]

        asm volatile("tensor_load_to_lds %0, %1" :: "s"(g0), "s"(g1) : "memory");
        __builtin_amdgcn_s_wait_tensorcnt(0);
    }
    __syncthreads();
#else
    {
        const float* mbase = model_out + (size_t)row0 * MODEL_W + NUM_NUM;
        const int TOT = TPB * D_CAT;
        for (int e = tid; e < TOT; e += TPB) {
            int rl = e / D_CAT;
            int c  = e - rl * D_CAT;
            const float* src = mbase + (size_t)rl * MODEL_W + c;
            uint32_t dst = (uint32_t)(uintptr_t)(&smem[rl * LSTRIDE + c]);
            asm volatile("global_load_async_to_lds_b32 %0, %1, off"
                         :: "v"(dst), "v"(src) : "memory");
        }
        asm volatile("s_wait_asynccnt 0" ::: "memory");
        __syncthreads();
    }
#endif

    const int r = row0 + tid;                  // this thread's row
    float* myrow = &smem[tid * LSTRIDE];       // model logits, later un_model
    const float* u_row = u_cat     + (size_t)r * D_CAT;
    const float* mo    = model_out + (size_t)r * MODEL_W;
    __builtin_prefetch(u_row,       0, 1);     // -> global_prefetch_b8
    __builtin_prefetch(u_row + 128, 0, 1);
    __builtin_prefetch(u_row + 256, 0, 1);

    // per-row schedule scalars
    const int tt = t_arr[r];
    const float lca    = tab[2 * T_STEPS + tt];
    const float l1mca  = tab[3 * T_STEPS + tt];
    const int   tm1    = (tt > 0) ? (tt - 1) : 0;
    const float lca1   = tab[2 * T_STEPS + tm1];
    const float l1mca1 = tab[3 * T_STEPS + tm1];
    const float lal    = tab[tt];
    const float l1ma   = tab[T_STEPS + tt];
    const bool  t0     = (tt == 0);

    // Gaussian term
    float lg = 0.f;
    const float* nz = noise + (size_t)r * NUM_NUM;
    #pragma unroll
    for (int i = 0; i < NUM_NUM; ++i) { float d = nz[i] - mo[i]; lg += d * d; }
    lg *= (1.0f / NUM_NUM);

    float kl = 0.f, dnll = 0.f;
    int off = 0;
    for (int f = 0; f < F_CAT; ++f) {
        const int   K    = c_K[f];
        const float logK = __logf((float)K);
        const int   xi   = x_cat_idx[(size_t)r * F_CAT + f];

        // q_pred(log_x_start, t): only two values per segment (one-hot input)
        const float evT = lae(lca,          l1mca - logK);
        const float evF = lae(LOGEPS + lca, l1mca - logK);

        // ---- pass 1: gumbel argmax sampling + online LSE of model logits ----
        int   samp = 0; float best = -3.4e38f;
        float rm = -3.4e38f, rs = 0.f;
        for (int j = 0; j < K; ++j) {
            float u = u_row[off + j];
            u = fminf(fmaxf(u, 1e-30f), 1.0f);
            float g  = -__logf(-__logf(u));
            float sc = ((j == xi) ? evT : evF) + g;
            if (sc >= best) { best = sc; samp = j; }
            float m  = myrow[off + j];
            float nm = fmaxf(rm, m);
            rs = rs * __expf(rm - nm) + __expf(m - nm);
            rm = nm;
        }
        const float Z0 = rm + __logf(rs);       // sliced_logsumexp(model_out_cat)

        // q_pred_one_timestep(log_x_t, t): two values per segment (one-hot)
        const float c1  = lae(lal,          l1ma - logK);   // sampled column
        const float c0v = lae(LOGEPS + lal, l1ma - logK);   // others

        // true-branch posterior: <=4 distinct unnormalized values per segment
        const float levT = t0 ? 0.f    : lae(lca1,          l1mca1 - logK);
        const float levF = t0 ? LOGEPS : lae(LOGEPS + lca1, l1mca1 - logK);
        const float uTT = levT + c1, uTF = levT + c0v;
        const float uFT = levF + c1, uFF = levF + c0v;
        float Zt;
        if (xi == samp) {
            Zt = lae(uTT, uFF + __logf((float)(K - 1)));
        } else {
            float third = (K > 2) ? (uFF + __logf((float)(K - 2))) : -3.4e38f;
            float m3 = fmaxf(fmaxf(uTF, uFT), third);
            Zt = m3 + __logf(__expf(uTF - m3) + __expf(uFT - m3) + __expf(third - m3));
        }

        // ---- pass 2: un_model (overwrites LDS) + online LSE ----
        rm = -3.4e38f; rs = 0.f;
        for (int j = 0; j < K; ++j) {
            float lx0 = myrow[off + j] - Z0;                       // log-softmax recon
            float lev = t0 ? lx0 : lae(lx0 + lca1, l1mca1 - logK); // q_pred(., t-1)
            float un  = lev + ((j == samp) ? c1 : c0v);
            myrow[off + j] = un;
            float nm = fmaxf(rm, un);
            rs = rs * __expf(rm - nm) + __expf(un - nm);
            rm = nm;
        }
        const float Zm = rm + __logf(rs);

        // true posterior case values (no per-column transcendentals needed)
        const float lTT = uTT - Zt, lTF = uTF - Zt, lFT = uFT - Zt, lFF = uFF - Zt;
        const float pTT = __expf(lTT), pTF = __expf(lTF);
        const float pFT = __expf(lFT), pFF = __expf(lFF);

        // ---- pass 3: KL and decoder NLL ----
        for (int j = 0; j < K; ++j) {
            float lmp = myrow[off + j] - Zm;                 // log_model_prob
            bool isT = (j == xi), isS = (j == samp);
            float ltp = isT ? (isS ? lTT : lTF) : (isS ? lFT : lFF);
            float p   = isT ? (isS ? pTT : pTF) : (isS ? pFT : pFF);
            kl += p * (ltp - lmp);
            if (isT) dnll -= lmp;
        }
        off += K;
    }

    const float klp = tab[4 * T_STEPS];
    const float lt  = t0 ? dnll : kl;
    out[r] = (lt * (float)T_STEPS + klp) * (1.0f / (float)F_CAT) + lg;
}

// ---------------- launcher ----------------
extern "C" void kernel_launch(void* const* d_in, const int* in_sizes, int n_in,
                              void* d_out, int out_size, void* d_ws, size_t ws_size,
                              hipStream_t stream) {
    (void)in_sizes; (void)n_in; (void)out_size; (void)ws_size;
    const int*   x_cat_idx = (const int*)  d_in[0];
    const int*   t_arr     = (const int*)  d_in[1];
    const float* noise     = (const float*)d_in[2];
    const float* u_cat     = (const float*)d_in[3];
    const float* model_out = (const float*)d_in[4];
    float*       out       = (float*)d_out;
    float*       ws        = (float*)d_ws;

    hipLaunchKernelGGL(gmd_init_tables, dim3(1), dim3(1024), 0, stream, ws);

    const size_t shmem = (size_t)TPB * LSTRIDE * sizeof(float);   // ~149 KB of 320 KB LDS
    hipLaunchKernelGGL(gmd_loss_kernel, dim3(B_ROWS / TPB), dim3(TPB), shmem, stream,
                       x_cat_idx, t_arr, noise, u_cat, model_out, ws, out);
}